// LiftProjectNetwork_26989574488300
// MI455X (gfx1250) — compile-verified
//
#include <hip/hip_runtime.h>
#include <math.h>

typedef __attribute__((ext_vector_type(2))) float v2f;
typedef __attribute__((ext_vector_type(8))) float v8f;

#define R 32          // feature width
#define KDIM 64       // concat(h,g) width

// ---------------------------------------------------------------- zero
__global__ void zero_kernel(float* __restrict__ p, int n) {
  int i = blockIdx.x * blockDim.x + threadIdx.x;
  if (i < n) p[i] = 0.0f;
}

// ---------------------------------------------------------------- scatter-add
// one wave per edge; lane = feature. Native fp32 global atomic.
__global__ void scatter_kernel(const float* __restrict__ x,
                               const long long* __restrict__ src,
                               const long long* __restrict__ dst,
                               const float* __restrict__ w,
                               float* __restrict__ g, int E) {
  int e = blockIdx.x * (blockDim.x >> 5) + (threadIdx.x >> 5);
  if (e >= E) return;
  int lane = threadIdx.x & 31;
  long long s = src[e];
  long long d = dst[e];
  float wt = w[e];
  float v = x[s * R + lane] * wt;
  unsafeAtomicAdd(&g[d * R + lane], v);   // -> global_atomic_add_f32
}

// ---------------------------------------------------------------- row L2-normalize (in place)
__global__ void l2norm_kernel(float* __restrict__ g, int N) {
  int n = blockIdx.x * (blockDim.x >> 5) + (threadIdx.x >> 5);
  if (n >= N) return;
  int lane = threadIdx.x & 31;
  long long idx = (long long)n * R + lane;
  float v = g[idx];
  float ss = v * v;
#pragma unroll
  for (int m = 16; m >= 1; m >>= 1) ss += __shfl_xor(ss, m, 32);
  float inv = 1.0f / fmaxf(sqrtf(ss), 1e-12f);
  g[idx] = v * inv;
}

// ---------------------------------------------------------------- fused GEMM + bias + activation
// out[16n x 32] = concat(h,g)[16n x 64] @ W[64 x 32] + b, then row-normalize (act=0) or tanh (act=1)
// One wave per 16-node tile, V_WMMA_F32_16X16X4_F32, 2 N-halves x 16 K-steps.
__global__ void __launch_bounds__(256)
gemm_kernel(const float* __restrict__ h, const float* __restrict__ g,
            const float* __restrict__ W, const float* __restrict__ b,
            float* __restrict__ out, int N, int act) {
  __shared__ float Wl[KDIM * R];
  __shared__ float bl[R];
  for (int i = threadIdx.x; i < KDIM * R; i += 256) Wl[i] = W[i];
  if (threadIdx.x < R) bl[threadIdx.x] = b[threadIdx.x];
  __syncthreads();

  int tile = blockIdx.x * 8 + (threadIdx.x >> 5);
  if (tile * 16 >= N) return;                  // whole-wave uniform: EXEC all-1s for WMMA
  int lane  = threadIdx.x & 31;
  int row   = lane & 15;                       // A-row / B-col index
  int khalf = lane >> 4;                       // K sub-pair selector (ISA A/B f32 layout)

  const float* hrow = h + ((long long)tile * 16 + row) * R;
  const float* grow = g + ((long long)tile * 16 + row) * R;

  v8f acc0 = {0.f,0.f,0.f,0.f,0.f,0.f,0.f,0.f};   // N = 0..15
  v8f acc1 = {0.f,0.f,0.f,0.f,0.f,0.f,0.f,0.f};   // N = 16..31

#pragma unroll
  for (int k = 0; k < 16; ++k) {
    int fb = k * 4 + khalf * 2;                // feature index within concat(h,g)
    v2f A;
    if (k < 8) { A.x = hrow[fb];      A.y = hrow[fb + 1]; }
    else       { A.x = grow[fb - 32]; A.y = grow[fb - 31]; }
    v2f B0, B1;
    B0.x = Wl[fb * R + row];        B0.y = Wl[(fb + 1) * R + row];
    B1.x = Wl[fb * R + 16 + row];   B1.y = Wl[(fb + 1) * R + 16 + row];
    acc0 = __builtin_amdgcn_wmma_f32_16x16x4_f32(false, A, false, B0,
                                                 (short)0, acc0, false, false);
    acc1 = __builtin_amdgcn_wmma_f32_16x16x4_f32(false, A, false, B1,
                                                 (short)0, acc1, false, false);
  }

  // bias: column n is fixed per lane per accumulator
  float b0 = bl[row], b1 = bl[16 + row];
#pragma unroll
  for (int i = 0; i < 8; ++i) { acc0[i] += b0; acc1[i] += b1; }

  if (act == 1) {
#pragma unroll
    for (int i = 0; i < 8; ++i) { acc0[i] = tanhf(acc0[i]); acc1[i] = tanhf(acc1[i]); }
  } else {
    // L2-normalize each node row in-register: node m = i + khalf*8 lives in
    // component i across the 16 lanes of this half -> xor masks 1,2,4,8.
#pragma unroll
    for (int i = 0; i < 8; ++i) {
      float ss = acc0[i] * acc0[i] + acc1[i] * acc1[i];
      ss += __shfl_xor(ss, 1, 32);
      ss += __shfl_xor(ss, 2, 32);
      ss += __shfl_xor(ss, 4, 32);
      ss += __shfl_xor(ss, 8, 32);
      float inv = 1.0f / fmaxf(sqrtf(ss), 1e-12f);
      acc0[i] *= inv; acc1[i] *= inv;
    }
  }

  long long base = (long long)tile * 16 * R;
#pragma unroll
  for (int i = 0; i < 8; ++i) {
    int m = i + khalf * 8;                     // C/D layout: M = vgpr (+8 upper half)
    out[base + m * R + row]      = acc0[i];
    out[base + m * R + 16 + row] = acc1[i];
  }
}

// ---------------------------------------------------------------- host orchestration
extern "C" void kernel_launch(void* const* d_in, const int* in_sizes, int n_in,
                              void* d_out, int out_size, void* d_ws, size_t ws_size,
                              hipStream_t stream) {
  const float*     x     = (const float*)d_in[0];
  const long long* ei    = (const long long*)d_in[1];   // int64 [2, E]
  const float*     ew    = (const float*)d_in[2];
  const float*     liftW = (const float*)d_in[3];       // [8, 64, 32]
  const float*     liftb = (const float*)d_in[4];       // [8, 32]
  const float*     projW = (const float*)d_in[5];       // [4, 64, 32]
  const float*     projb = (const float*)d_in[6];       // [4, 32]

  int N = in_sizes[0] / R;
  int E = in_sizes[2];
  const long long* src = ei;
  const long long* dst = ei + E;

  float* gbuf = (float*)d_ws;
  float* hA   = gbuf + (size_t)N * R;
  float* hB   = hA   + (size_t)N * R;
  float* outp = (float*)d_out;

  int nElem = N * R;
  int zgrid = (nElem + 255) / 256;
  int egrid = (E + 7) / 8;
  int ngrid = (N + 7) / 8;
  int tgrid = ((N + 15) / 16 + 7) / 8;

  const float* hcur = x;
  for (int layer = 0; layer < 12; ++layer) {
    zero_kernel<<<zgrid, 256, 0, stream>>>(gbuf, nElem);
    scatter_kernel<<<egrid, 256, 0, stream>>>(hcur, src, dst, ew, gbuf, E);

    const float *Wp, *bp;
    int act;
    if (layer < 8) {
      l2norm_kernel<<<ngrid, 256, 0, stream>>>(gbuf, N);
      Wp = liftW + (size_t)layer * KDIM * R;
      bp = liftb + (size_t)layer * R;
      act = 0;   // normalize output
    } else {
      Wp = projW + (size_t)(layer - 8) * KDIM * R;
      bp = projb + (size_t)(layer - 8) * R;
      act = 1;   // tanh output
    }

    float* hnext = (layer == 11) ? outp : ((layer & 1) ? hB : hA);
    gemm_kernel<<<tgrid, 256, 0, stream>>>(hcur, gbuf, Wp, bp, hnext, N, act);
    hcur = hnext;
  }
}